// CuGraphRelGraphConv_85057532330307
// MI455X (gfx1250) — compile-verified
//
#include <hip/hip_runtime.h>

#define DIN   128
#define DOUT  128
#define RNUM  64
#define BNUM  8
#define KTOT  ((BNUM + 1) * DIN)   // 1152
#define KCHUNK 64
#define NPAIR  (KCHUNK / 2)        // 32 K-pairs per chunk
#define PSTRIDE 144                // float2 stride per K-pair row (16 mod 32)

typedef float v2f __attribute__((ext_vector_type(2)));
typedef float v8f __attribute__((ext_vector_type(8)));

// ---------------- zero workspace (deg counters + h_agg accumulator) ---------
__global__ void rgcn_zero_kernel(uint4* __restrict__ ws, long n4) {
  long i = (long)blockIdx.x * blockDim.x + threadIdx.x;
  long stride = (long)gridDim.x * blockDim.x;
  uint4 z = {0u, 0u, 0u, 0u};
  for (; i < n4; i += stride) ws[i] = z;
}

// ---------------- per-(dst, rel) degree --------------------------------------
__global__ void rgcn_deg_kernel(const int* __restrict__ dst,
                                const int* __restrict__ et,
                                unsigned* __restrict__ deg, int E) {
  int e = blockIdx.x * blockDim.x + threadIdx.x;
  if (e < E) atomicAdd(&deg[(long)dst[e] * RNUM + et[e]], 1u);
}

// ---------------- basis-weighted edge scatter --------------------------------
// one wave32 per edge; lane handles 4 channels (float4); 8 bases -> 32 atomics
__global__ void rgcn_scatter_kernel(const float* __restrict__ feat,
                                    const float* __restrict__ coeff,
                                    const int* __restrict__ src,
                                    const int* __restrict__ dst,
                                    const int* __restrict__ et,
                                    const unsigned* __restrict__ deg,
                                    float* __restrict__ hagg, int E) {
  int e = blockIdx.x * (blockDim.x >> 5) + (threadIdx.x >> 5);
  if (e >= E) return;
  int lane = threadIdx.x & 31;
  int s = src[e], d = dst[e], t = et[e];
  float dg = (float)deg[(long)d * RNUM + t];
  float norm = 1.0f / fmaxf(dg, 1.0f);
  float4 x = *(const float4*)(feat + (long)s * DIN + lane * 4);
  x.x *= norm; x.y *= norm; x.z *= norm; x.w *= norm;
  float* base = hagg + (long)d * (BNUM * DIN) + lane * 4;
#pragma unroll
  for (int b = 0; b < BNUM; ++b) {
    float c = coeff[t * BNUM + b];
    float* p = base + b * DIN;
    unsafeAtomicAdd(p + 0, c * x.x);
    unsafeAtomicAdd(p + 1, c * x.y);
    unsafeAtomicAdd(p + 2, c * x.z);
    unsafeAtomicAdd(p + 3, c * x.w);
  }
}

// ---------------- dense GEMM via f32 WMMA ------------------------------------
// out[M=Nn,128] = concat(hagg[M,1024], feat[M,128]) @ W[1152,128] + bias
// workgroup: 256 threads = 8 waves; tile 64 rows x 128 cols
// wave w: rowTile = w>>1 (16 rows), colBase = (w&1)*64 (4 x 16-col WMMA tiles)
// LDS holds W chunk K-pair-interleaved: lds[(p, n)] = {W[k0+2p][n], W[k0+2p+1][n]}
// so a B fragment (b.x=W[klo][n], b.y=W[klo+1][n]) is a single b64 load.
__global__ __launch_bounds__(256)
void rgcn_gemm_kernel(const float* __restrict__ hagg,
                      const float* __restrict__ feat,
                      const float* __restrict__ W,      // [(B+1)*DIN, DOUT]
                      const float* __restrict__ bias,
                      float* __restrict__ out, int Nn) {
  __shared__ v2f ldsW[NPAIR * PSTRIDE];                 // 36.9 KB

  int tid = threadIdx.x;
  int wid = tid >> 5;
  int lane = tid & 31;
  int halflane = lane >> 4;        // 0: lanes 0-15, 1: lanes 16-31
  int l16 = lane & 15;

  int rowTile = wid >> 1;          // 0..3
  int colBase = (wid & 1) * 64;    // 0 or 64
  int rowBlock = blockIdx.x * 64;

  int row0 = rowBlock + rowTile * 16 + l16;       // A-fragment row (M = l16)
  int rowA = row0 < Nn ? row0 : (Nn - 1);         // clamp: EXEC must stay full

  v8f acc[4] = {v8f{}, v8f{}, v8f{}, v8f{}};

  for (int k0 = 0; k0 < KTOT; k0 += KCHUNK) {
    // cooperative interleaved stage: (p, c4) pairs, 1024 total / 256 threads
#pragma unroll
    for (int i = 0; i < 4; ++i) {
      int id = tid + i * 256;                     // 0..1023
      int p  = id >> 5;                           // K-pair 0..31
      int c4 = id & 31;                           // float4 column group
      float4 r0 = *(const float4*)(W + (long)(k0 + 2 * p)     * DOUT + c4 * 4);
      float4 r1 = *(const float4*)(W + (long)(k0 + 2 * p + 1) * DOUT + c4 * 4);
      float4* dstp = (float4*)&ldsW[p * PSTRIDE + c4 * 4];
      dstp[0] = make_float4(r0.x, r1.x, r0.y, r1.y);
      dstp[1] = make_float4(r0.z, r1.z, r0.w, r1.w);
    }
    __syncthreads();

    // A source: first 1024 K come from hagg, last 128 from feat (self-loop)
    const float* Abase = (k0 < BNUM * DIN)
        ? hagg + (long)rowA * (BNUM * DIN) + k0
        : feat + (long)rowA * DIN + (k0 - BNUM * DIN);

#pragma unroll 4
    for (int kk = 0; kk < KCHUNK; kk += 4) {
      // A 16x4 f32 fragment: lanes 0-15 hold K=kk,kk+1; lanes 16-31 K=kk+2,kk+3
      int klo = kk + (halflane << 1);
      v2f a = *(const v2f*)(Abase + klo);
      int p = (kk >> 1) + halflane;               // K-pair index for B fragment
#pragma unroll
      for (int t = 0; t < 4; ++t) {
        int n = colBase + t * 16 + l16;
        v2f b = ldsW[p * PSTRIDE + n];            // single ds_load_b64
        acc[t] = __builtin_amdgcn_wmma_f32_16x16x4_f32(
            false, a, false, b, (short)0, acc[t], false, false);
      }
    }
    __syncthreads();
  }

  // C/D layout: VGPR i -> M = i + 8*halflane, N = l16
#pragma unroll
  for (int t = 0; t < 4; ++t) {
    int n = colBase + t * 16 + l16;
    float bv = bias[n];
#pragma unroll
    for (int i = 0; i < 8; ++i) {
      int row = rowBlock + rowTile * 16 + i + (halflane << 3);
      if (row < Nn) out[(long)row * DOUT + n] = acc[t][i] + bv;
    }
  }
}

// ---------------------------------------------------------------------------
extern "C" void kernel_launch(void* const* d_in, const int* in_sizes, int n_in,
                              void* d_out, int out_size, void* d_ws, size_t ws_size,
                              hipStream_t stream) {
  const float* feat  = (const float*)d_in[0];
  const float* W     = (const float*)d_in[1];
  const float* coeff = (const float*)d_in[2];
  const float* bias  = (const float*)d_in[3];
  const int*   src   = (const int*)d_in[4];
  const int*   dst   = (const int*)d_in[5];
  const int*   et    = (const int*)d_in[6];
  float* out = (float*)d_out;

  int Nn = in_sizes[0] / DIN;       // 50000
  int E  = in_sizes[4];             // 1600000

  // workspace layout: [deg: Nn*R u32][h_agg: Nn*B*DIN f32], deg region 256B-aligned
  size_t degBytes  = ((size_t)Nn * RNUM * sizeof(unsigned) + 255) & ~(size_t)255;
  unsigned* deg = (unsigned*)d_ws;
  float* hagg = (float*)((char*)d_ws + degBytes);
  size_t totalBytes = degBytes + (size_t)Nn * BNUM * DIN * sizeof(float);

  // 1) zero counters + accumulators
  long n4 = (long)((totalBytes + 15) / 16);
  rgcn_zero_kernel<<<8192, 256, 0, stream>>>((uint4*)d_ws, n4);

  // 2) per-(dst, rel) degree
  rgcn_deg_kernel<<<(E + 255) / 256, 256, 0, stream>>>(dst, et, deg, E);

  // 3) edge scatter: 8 edges (waves) per 256-thread block
  rgcn_scatter_kernel<<<(E + 7) / 8, 256, 0, stream>>>(
      feat, coeff, src, dst, et, deg, hagg, E);

  // 4) WMMA GEMM + bias
  rgcn_gemm_kernel<<<(Nn + 63) / 64, 256, 0, stream>>>(
      hagg, feat, W, bias, out, Nn);
}